// GAT_13048110645524
// MI455X (gfx1250) — compile-verified
//
#include <hip/hip_runtime.h>
#include <math.h>

// ---------------------------------------------------------------------------
// Types for CDNA5 WMMA (f32 16x16x4: A/B = 2 VGPRs/lane, C/D = 8 VGPRs/lane)
// ---------------------------------------------------------------------------
typedef __attribute__((ext_vector_type(2))) float v2f;
typedef __attribute__((ext_vector_type(8))) float v8f;

#define NEG_SLOPE 0.2f

// ---------------------------------------------------------------------------
// GEMM: C[M x ncols] = A[M x 256] * B[256 x ncols]   (K fixed at 256)
// Block = 256 threads = 8 waves. Block tile = 16 rows x 128 cols.
// Each wave computes one 16x16 tile via V_WMMA_F32_16X16X4_F32, K-loop of 64.
// A staged fully in LDS (16x256, stride 260 to avoid bank conflicts),
// B staged in K-chunks of 32 (32x128, stride 132).
// ---------------------------------------------------------------------------
__global__ __launch_bounds__(256) void gemm_wmma_f32(
    const float* __restrict__ A, const float* __restrict__ B,
    float* __restrict__ C, int ncols)
{
    const int K = 256;
    __shared__ float As[16 * 260];
    __shared__ float Bs[32 * 132];

    const int tid  = threadIdx.x;
    const int wave = tid >> 5;
    const int lane = tid & 31;
    const int rowBase = blockIdx.x * 16;
    const int colBase = blockIdx.y * 128;

    // Stage full A tile (16 x 256), coalesced.
    for (int i = tid; i < 16 * 256; i += 256) {
        int r = i >> 8, k = i & 255;
        As[r * 260 + k] = A[(size_t)(rowBase + r) * K + k];
    }

    v8f acc = {};
    const int row  = lane & 15;          // A row owned by this lane
    const int koff = (lane >> 4) * 2;    // lanes 16..31 hold K+2,K+3
    const int col  = lane & 15;          // B/D column owned by this lane

    for (int kb = 0; kb < K; kb += 32) {
        __syncthreads();
        // Stage B chunk (32 x 128), coalesced.
        for (int i = tid; i < 32 * 128; i += 256) {
            int kr = i >> 7, c = i & 127;
            Bs[kr * 132 + c] = B[(size_t)(kb + kr) * ncols + colBase + c];
        }
        __syncthreads();

#pragma unroll
        for (int kk = 0; kk < 32; kk += 4) {
            v2f a, b;
            a.x = As[row * 260 + kb + kk + koff];
            a.y = As[row * 260 + kb + kk + koff + 1];
            b.x = Bs[(kk + koff) * 132 + wave * 16 + col];
            b.y = Bs[(kk + koff + 1) * 132 + wave * 16 + col];
            acc = __builtin_amdgcn_wmma_f32_16x16x4_f32(
                false, a, false, b, (short)0, acc, false, false);
        }
    }

    // D layout: VGPR v -> lanes 0-15: (M=v, N=lane); lanes 16-31: (M=v+8, N=lane-16)
#pragma unroll
    for (int v = 0; v < 8; v++) {
        int mrow = v + ((lane >> 4) << 3);
        C[(size_t)(rowBase + mrow) * ncols + colBase + wave * 16 + col] = acc[v];
    }
}

// ---------------------------------------------------------------------------
// Per-node attention projections: el[n,h] = <ft[n,h,:], attn_l[h,:]>, same er.
// One wave per node. Lane l owns C/32 contiguous features => exactly one head.
// ---------------------------------------------------------------------------
template <int H, int D>
__global__ __launch_bounds__(256) void attn_proj(
    const float* __restrict__ ft,
    const float* __restrict__ attn_l, const float* __restrict__ attn_r,
    float* __restrict__ el, float* __restrict__ er, int n)
{
    const int C = H * D;
    const int PER = C / 32;
    const int GROUP = D / PER;   // lanes per head group

    int wid  = (blockIdx.x * blockDim.x + threadIdx.x) >> 5;
    int lane = threadIdx.x & 31;
    if (wid >= n) return;

    const int base = lane * PER;
    const int head = base / D;

    const float4* fr = (const float4*)(ft + (size_t)wid * C + base);
    const float4* al = (const float4*)(attn_l + base);
    const float4* ar = (const float4*)(attn_r + base);

    float sl = 0.f, sr = 0.f;
#pragma unroll
    for (int q = 0; q < PER / 4; q++) {
        float4 v = fr[q], l4 = al[q], r4 = ar[q];
        sl += v.x * l4.x + v.y * l4.y + v.z * l4.z + v.w * l4.w;
        sr += v.x * r4.x + v.y * r4.y + v.z * r4.z + v.w * r4.w;
    }
#pragma unroll
    for (int o = 1; o < GROUP; o <<= 1) {
        sl += __shfl_xor(sl, o, 32);
        sr += __shfl_xor(sr, o, 32);
    }
    if ((lane & (GROUP - 1)) == 0) {
        el[(size_t)wid * H + head] = sl;
        er[(size_t)wid * H + head] = sr;
    }
}

// ---------------------------------------------------------------------------
// CSR construction helpers
// ---------------------------------------------------------------------------
__global__ void zero_ints(int* __restrict__ p, int n)
{
    int i = blockIdx.x * blockDim.x + threadIdx.x;
    if (i < n) p[i] = 0;
}

__global__ void count_deg(const int* __restrict__ dst, int* __restrict__ deg, int e)
{
    int i = blockIdx.x * blockDim.x + threadIdx.x;
    if (i < e) atomicAdd(&deg[dst[i]], 1);
}

// Single-block exclusive scan over n (<= 1024*chunk) elements.
__global__ __launch_bounds__(1024) void scan_kernel(
    const int* __restrict__ deg, int* __restrict__ offs, int n)
{
    __shared__ int part[1024];
    const int tid = threadIdx.x;
    const int CH  = (n + 1023) / 1024;
    const int base = tid * CH;

    int s = 0;
    for (int j = 0; j < CH; j++) {
        int idx = base + j;
        if (idx < n) s += deg[idx];
    }
    part[tid] = s;
    __syncthreads();
    for (int off = 1; off < 1024; off <<= 1) {
        int v = (tid >= off) ? part[tid - off] : 0;
        __syncthreads();
        part[tid] += v;
        __syncthreads();
    }
    int run = (tid == 0) ? 0 : part[tid - 1];
    for (int j = 0; j < CH; j++) {
        int idx = base + j;
        if (idx < n) { offs[idx] = run; run += deg[idx]; }
    }
    if (tid == 0) offs[n] = part[1023];
}

__global__ void scatter_edges(const int* __restrict__ src, const int* __restrict__ dst,
                              const int* __restrict__ offs, int* __restrict__ cursor,
                              int* __restrict__ csr_src, int e)
{
    int i = blockIdx.x * blockDim.x + threadIdx.x;
    if (i < e) {
        int d = dst[i];
        int pos = atomicAdd(&cursor[d], 1);
        csr_src[offs[d] + pos] = src[i];
    }
}

// ---------------------------------------------------------------------------
// Fused edge-softmax + aggregation. One wave per dst node.
// Pass 1: segment max of leaky(el[src]+er[dst]) per head.
// Pass 2: exp-sum + register accumulation of ft[src]*ex (coalesced float4).
// Epilogue: normalize, +bias, optional ReLU. Empty segments -> bias only.
// ---------------------------------------------------------------------------
template <int H, int D, bool RELU>
__global__ __launch_bounds__(256) void aggregate(
    const float* __restrict__ ft,
    const float* __restrict__ el, const float* __restrict__ er,
    const int* __restrict__ offs, const int* __restrict__ csr_src,
    const float* __restrict__ bias,
    float* __restrict__ out, int n)
{
    const int C = H * D;
    const int PER = C / 32;

    int wid  = (blockIdx.x * blockDim.x + threadIdx.x) >> 5;
    int lane = threadIdx.x & 31;
    if (wid >= n) return;

    const int head = (lane * PER) / D;
    const float erd = er[(size_t)wid * H + head];
    const int s0 = offs[wid];
    const int s1 = offs[wid + 1];

    // Pass 1: segment max (per head; each head's 32/H lanes compute it redundantly)
    float m = -3.402823466e38f;
    for (int i = s0; i < s1; i++) {
        int s = csr_src[i];
        float e = el[(size_t)s * H + head] + erd;
        e = (e > 0.f) ? e : NEG_SLOPE * e;
        m = fmaxf(m, e);
    }

    // Pass 2: exp-sum + accumulate ft[src] * ex
    float acc[PER];
#pragma unroll
    for (int j = 0; j < PER; j++) acc[j] = 0.f;
    float sum = 0.f;

    for (int i = s0; i < s1; i++) {
        int s = csr_src[i];
        float e = el[(size_t)s * H + head] + erd;
        e = (e > 0.f) ? e : NEG_SLOPE * e;
        float ex = __expf(e - m);
        sum += ex;
        const float4* fr = (const float4*)(ft + (size_t)s * C + lane * PER);
#pragma unroll
        for (int q = 0; q < PER / 4; q++) {
            float4 v = fr[q];
            acc[q * 4 + 0] += v.x * ex;
            acc[q * 4 + 1] += v.y * ex;
            acc[q * 4 + 2] += v.z * ex;
            acc[q * 4 + 3] += v.w * ex;
        }
    }

    float r = (s1 > s0) ? (1.0f / sum) : 0.0f;
#pragma unroll
    for (int j = 0; j < PER; j++) {
        float v = acc[j] * r + bias[lane * PER + j];
        if (RELU) v = fmaxf(v, 0.f);
        out[(size_t)wid * C + lane * PER + j] = v;
    }
}

// ---------------------------------------------------------------------------
// Host-side launcher
// ---------------------------------------------------------------------------
extern "C" void kernel_launch(void* const* d_in, const int* in_sizes, int n_in,
                              void* d_out, int out_size, void* d_ws, size_t ws_size,
                              hipStream_t stream)
{
    const float* feat    = (const float*)d_in[0];
    const int*   src     = (const int*)d_in[1];
    const int*   dst     = (const int*)d_in[2];
    const float* W1      = (const float*)d_in[3];
    const float* attn_l1 = (const float*)d_in[4];
    const float* attn_r1 = (const float*)d_in[5];
    const float* bias1   = (const float*)d_in[6];
    const float* W2      = (const float*)d_in[7];
    const float* attn_l2 = (const float*)d_in[8];
    const float* attn_r2 = (const float*)d_in[9];
    const float* bias2   = (const float*)d_in[10];

    const int N = in_sizes[0] / 256;   // 50000
    const int E = in_sizes[1];         // 800000

    // Workspace carve-up
    char* w = (char*)d_ws;
    float* ft = (float*)w;  w += (size_t)N * 256 * sizeof(float);   // ft1, reused as ft2
    float* h  = (float*)w;  w += (size_t)N * 256 * sizeof(float);   // layer-1 output
    float* el = (float*)w;  w += (size_t)N * 4 * sizeof(float);
    float* er = (float*)w;  w += (size_t)N * 4 * sizeof(float);
    int* offs = (int*)w;    w += (size_t)(N + 1) * sizeof(int);
    int* cnt  = (int*)w;    w += (size_t)N * sizeof(int);
    int* csr  = (int*)w;    w += (size_t)E * sizeof(int);

    const int rowTiles   = N / 16;                 // 3125 (N divisible by 16)
    const int nodeBlocks = (N * 32 + 255) / 256;   // one wave per node
    const int eBlocks    = (E + 255) / 256;
    const int nBlocks    = (N + 255) / 256;

    // ---- CSR build (shared by both layers) ----
    zero_ints<<<nBlocks, 256, 0, stream>>>(cnt, N);
    count_deg<<<eBlocks, 256, 0, stream>>>(dst, cnt, E);
    scan_kernel<<<1, 1024, 0, stream>>>(cnt, offs, N);
    zero_ints<<<nBlocks, 256, 0, stream>>>(cnt, N);
    scatter_edges<<<eBlocks, 256, 0, stream>>>(src, dst, offs, cnt, csr, E);

    // ---- Layer 1: H=4, D=64, ReLU ----
    gemm_wmma_f32<<<dim3(rowTiles, 2), 256, 0, stream>>>(feat, W1, ft, 256);
    attn_proj<4, 64><<<nodeBlocks, 256, 0, stream>>>(ft, attn_l1, attn_r1, el, er, N);
    aggregate<4, 64, true><<<nodeBlocks, 256, 0, stream>>>(
        ft, el, er, offs, csr, bias1, h, N);

    // ---- Layer 2: H=1, D=128, no ReLU ----
    gemm_wmma_f32<<<dim3(rowTiles, 1), 256, 0, stream>>>(h, W2, ft, 128);
    attn_proj<1, 128><<<nodeBlocks, 256, 0, stream>>>(ft, attn_l2, attn_r2, el, er, N);
    aggregate<1, 128, false><<<nodeBlocks, 256, 0, stream>>>(
        ft, el, er, offs, csr, bias2, (float*)d_out, N);
}